// PointNetSetAbstraction_86268713107581
// MI455X (gfx1250) — compile-verified
//
#include <hip/hip_runtime.h>
#include <hip/hip_bf16.h>
#include <math.h>

#define B_      16
#define N_      4096
#define C_      64
#define NPOINT  1024
#define NSAMPLE 32
#define PITCH   68      // 32x68 fp32 activation tiles in LDS; 68%64=4 -> conflict-free
#define BN_EPS  1e-5f

typedef float v2f __attribute__((ext_vector_type(2)));
typedef float v8f __attribute__((ext_vector_type(8)));

// D(16x16,f32) = A(16x4,f32) * B(4x16,f32) + C   — CDNA5 fp32 WMMA
static __device__ inline v8f wmma4(v2f a, v2f b, v8f c) {
    return __builtin_amdgcn_wmma_f32_16x16x4_f32(
        /*neg_a=*/false, a, /*neg_b=*/false, b,
        /*c_mod=*/(short)0, c, /*reuse_a=*/false, /*reuse_b=*/false);
}

// ---------------------------------------------------------------- fuse BN into weights
// Weights are stored K-pair-packed: Wp[kpair*COUT + o] = {W'[2kp][o], W'[2kp+1][o]}
// so each WMMA B-tile is a single global b64 load per lane.
__global__ __launch_bounds__(256) void fuse_weights_kernel(
    const float* __restrict__ W0, const float* __restrict__ b0, const float* __restrict__ g0,
    const float* __restrict__ be0, const float* __restrict__ m0, const float* __restrict__ v0,
    const float* __restrict__ W1, const float* __restrict__ b1, const float* __restrict__ g1,
    const float* __restrict__ be1, const float* __restrict__ m1, const float* __restrict__ v1,
    const float* __restrict__ W2, const float* __restrict__ b2, const float* __restrict__ g2,
    const float* __restrict__ be2, const float* __restrict__ m2, const float* __restrict__ v2_,
    v2f* __restrict__ Wp0, float* __restrict__ b0f,
    v2f* __restrict__ Wp1, float* __restrict__ b1f,
    v2f* __restrict__ Wp2, float* __restrict__ b2f)
{
    const int tid = threadIdx.x;
    // layer0: W0 [64,67] -> Wp0 [34 kpairs][64], zero-pad K=67, BN scale folded
    for (int i = tid; i < 34 * 64; i += 256) {
        int o = i & 63, kp = i >> 6;
        float s = g0[o] / sqrtf(v0[o] + BN_EPS);
        int k0 = 2 * kp, k1 = 2 * kp + 1;
        v2f w;
        w.x = (k0 < 67) ? W0[o * 67 + k0] * s : 0.0f;
        w.y = (k1 < 67) ? W0[o * 67 + k1] * s : 0.0f;
        Wp0[i] = w;
    }
    for (int i = tid; i < 64; i += 256) {
        float s = g0[i] / sqrtf(v0[i] + BN_EPS);
        b0f[i] = (b0[i] - m0[i]) * s + be0[i];
    }
    // layer1: W1 [64,64] -> Wp1 [32][64]
    for (int i = tid; i < 32 * 64; i += 256) {
        int o = i & 63, kp = i >> 6;
        float s = g1[o] / sqrtf(v1[o] + BN_EPS);
        v2f w;
        w.x = W1[o * 64 + 2 * kp + 0] * s;
        w.y = W1[o * 64 + 2 * kp + 1] * s;
        Wp1[i] = w;
    }
    for (int i = tid; i < 64; i += 256) {
        float s = g1[i] / sqrtf(v1[i] + BN_EPS);
        b1f[i] = (b1[i] - m1[i]) * s + be1[i];
    }
    // layer2: W2 [128,64] -> Wp2 [32][128]
    for (int i = tid; i < 32 * 128; i += 256) {
        int o = i & 127, kp = i >> 7;
        float s = g2[o] / sqrtf(v2_[o] + BN_EPS);
        v2f w;
        w.x = W2[o * 64 + 2 * kp + 0] * s;
        w.y = W2[o * 64 + 2 * kp + 1] * s;
        Wp2[i] = w;
    }
    for (int i = tid; i < 128; i += 256) {
        float s = g2[i] / sqrtf(v2_[i] + BN_EPS);
        b2f[i] = (b2[i] - m2[i]) * s + be2[i];
    }
}

// ---------------------------------------------------------------- farthest point sampling
__global__ __launch_bounds__(256) void fps_kernel(const float* __restrict__ xyz,
                                                  float* __restrict__ new_xyz)
{
    __shared__ float sx[N_], sy[N_], sz[N_];
    __shared__ float red_val[8];
    __shared__ int   red_idx[8];
    __shared__ int   s_last;
    const int b   = blockIdx.x;
    const int tid = threadIdx.x;
    const float* bx = xyz + (size_t)b * N_ * 3;

    float dloc[16];
#pragma unroll
    for (int i = 0; i < 16; ++i) {
        int p = tid + 256 * i;
        sx[p] = bx[p * 3 + 0];
        sy[p] = bx[p * 3 + 1];
        sz[p] = bx[p * 3 + 2];
        dloc[i] = 1e10f;
    }
    if (tid == 0) {
        new_xyz[(b * NPOINT + 0) * 3 + 0] = bx[0];
        new_xyz[(b * NPOINT + 0) * 3 + 1] = bx[1];
        new_xyz[(b * NPOINT + 0) * 3 + 2] = bx[2];
    }
    __syncthreads();

    int last = 0;
    for (int it = 1; it < NPOINT; ++it) {
        const float lx = sx[last], ly = sy[last], lz = sz[last];
        float val = -1.0f;
        int   best = tid;
#pragma unroll
        for (int i = 0; i < 16; ++i) {
            int p = tid + 256 * i;
            float dx = sx[p] - lx, dy = sy[p] - ly, dz = sz[p] - lz;
            float dist = dx * dx + dy * dy + dz * dz;
            dloc[i] = fminf(dloc[i], dist);
            if (dloc[i] > val) { val = dloc[i]; best = p; }   // strict > keeps smallest p
        }
        // wave32 argmax reduction (no barriers); first-max tie-break = smallest index
#pragma unroll
        for (int off = 16; off > 0; off >>= 1) {
            float ov = __shfl_xor(val, off, 32);
            int   oi = __shfl_xor(best, off, 32);
            if (ov > val || (ov == val && oi < best)) { val = ov; best = oi; }
        }
        if ((tid & 31) == 0) { red_val[tid >> 5] = val; red_idx[tid >> 5] = best; }
        __syncthreads();
        if (tid == 0) {
            float bv = red_val[0];
            int   bi = red_idx[0];
#pragma unroll
            for (int w = 1; w < 8; ++w) {
                float wv = red_val[w];
                int   wi = red_idx[w];
                if (wv > bv || (wv == bv && wi < bi)) { bv = wv; bi = wi; }
            }
            s_last = bi;
            new_xyz[(b * NPOINT + it) * 3 + 0] = sx[bi];
            new_xyz[(b * NPOINT + it) * 3 + 1] = sy[bi];
            new_xyz[(b * NPOINT + it) * 3 + 2] = sz[bi];
        }
        __syncthreads();
        last = s_last;
    }
}

// ---------------------------------------------------------------- top-32 nearest neighbors
#define KNN_T 128
#define TILE  512
__global__ __launch_bounds__(KNN_T) void knn_kernel(const float* __restrict__ xyz,
                                                    const float* __restrict__ new_xyz,
                                                    int* __restrict__ gidx)
{
    __shared__ float tx[TILE], ty[TILE], tz[TILE];
    __shared__ float ld[NSAMPLE * KNN_T];   // per-thread top-32 dists, thread-strided
    __shared__ int   li[NSAMPLE * KNN_T];
    const int b    = blockIdx.x >> 3;
    const int g    = blockIdx.x & 7;
    const int tid  = threadIdx.x;
    const int cent = g * KNN_T + tid;

    const float cx = new_xyz[(b * NPOINT + cent) * 3 + 0];
    const float cy = new_xyz[(b * NPOINT + cent) * 3 + 1];
    const float cz = new_xyz[(b * NPOINT + cent) * 3 + 2];

#pragma unroll
    for (int e = 0; e < NSAMPLE; ++e) ld[e * KNN_T + tid] = 3.4e38f;
    float curmax = 3.4e38f;
    int   curpos = 0;
    const float* bx = xyz + (size_t)b * N_ * 3;

    for (int t = 0; t < N_ / TILE; ++t) {
        __syncthreads();
#pragma unroll
        for (int i = 0; i < TILE / KNN_T; ++i) {
            int p = t * TILE + i * KNN_T + tid;
            tx[i * KNN_T + tid] = bx[p * 3 + 0];
            ty[i * KNN_T + tid] = bx[p * 3 + 1];
            tz[i * KNN_T + tid] = bx[p * 3 + 2];
        }
        __syncthreads();
        for (int j = 0; j < TILE; ++j) {
            float dx = tx[j] - cx, dy = ty[j] - cy, dz = tz[j] - cz;
            float d2 = dx * dx + dy * dy + dz * dz;
            if (d2 < curmax) {
                ld[curpos * KNN_T + tid] = d2;
                li[curpos * KNN_T + tid] = t * TILE + j;
                curmax = -1.0f;
#pragma unroll
                for (int e = 0; e < NSAMPLE; ++e) {
                    float v = ld[e * KNN_T + tid];
                    if (v > curmax) { curmax = v; curpos = e; }
                }
            }
        }
    }
    int* go = gidx + (size_t)(b * NPOINT + cent) * NSAMPLE;
#pragma unroll
    for (int e = 0; e < NSAMPLE; ++e) go[e] = li[e * KNN_T + tid];
}

// ---------------------------------------------------------------- WMMA MLP layer
template <int KSTEPS, int NT, int COUT>
static __device__ inline void layer_gemm(const float* in_lds, float* out_lds,
                                         const v2f* __restrict__ Wp,
                                         const float* __restrict__ bias, int lane)
{
    v8f acc[2][NT];
    v8f zero = {};
#pragma unroll
    for (int mt = 0; mt < 2; ++mt)
#pragma unroll
        for (int n = 0; n < NT; ++n) acc[mt][n] = zero;

    const int mrow  = lane & 15;
    const int khalf = (lane >> 4) * 2;
    const int ncol  = lane & 15;

    for (int k = 0; k < KSTEPS; ++k) {
        const int kc = 4 * k + khalf;                                // even
        v2f a0 = *(const v2f*)(in_lds + mrow * PITCH + kc);          // M rows 0..15
        v2f a1 = *(const v2f*)(in_lds + (mrow + 16) * PITCH + kc);   // M rows 16..31
#pragma unroll
        for (int n = 0; n < NT; ++n) {
            v2f bt = Wp[(kc >> 1) * COUT + n * 16 + ncol];           // one b64 load
            acc[0][n] = wmma4(a0, bt, acc[0][n]);
            acc[1][n] = wmma4(a1, bt, acc[1][n]);
        }
    }
    const int rbase = (lane >> 4) * 8;
#pragma unroll
    for (int n = 0; n < NT; ++n) {
        float bb = bias[n * 16 + ncol];
#pragma unroll
        for (int mt = 0; mt < 2; ++mt)
#pragma unroll
            for (int j = 0; j < 8; ++j) {
                int rrow = mt * 16 + rbase + j;
                out_lds[rrow * PITCH + n * 16 + ncol] = fmaxf(acc[mt][n][j] + bb, 0.0f);
            }
    }
}

// ---------------------------------------------------------------- gather + MLP + max-pool
__global__ __launch_bounds__(32) void mlp_kernel(
    const float* __restrict__ xyz, const float* __restrict__ features,
    const int* __restrict__ gidx, const float* __restrict__ new_xyz,
    float* __restrict__ out_feat,
    const v2f* __restrict__ Wp0, const float* __restrict__ b0f,
    const v2f* __restrict__ Wp1, const float* __restrict__ b1f,
    const v2f* __restrict__ Wp2, const float* __restrict__ b2f)
{
    __shared__ __align__(16) float bufA[32 * PITCH];
    __shared__ __align__(16) float bufB[32 * PITCH];
    const int cg   = blockIdx.x;        // b*1024 + p
    const int b    = cg >> 10;
    const int lane = threadIdx.x;

    // gather: row `lane` = sample `lane`, 3 centered xyz + 64 feature channels + zero pad
    const int   idx = gidx[(size_t)cg * NSAMPLE + lane];
    const float cx = new_xyz[cg * 3 + 0];
    const float cy = new_xyz[cg * 3 + 1];
    const float cz = new_xyz[cg * 3 + 2];
    const float* pt = xyz + ((size_t)b * N_ + idx) * 3;
    float* row = bufA + lane * PITCH;
    row[0] = pt[0] - cx;
    row[1] = pt[1] - cy;
    row[2] = pt[2] - cz;
    const float* fb = features + (size_t)b * C_ * N_ + idx;
#pragma unroll 8
    for (int c = 0; c < C_; ++c) row[3 + c] = fb[(size_t)c * N_];
    row[67] = 0.0f;
    __syncthreads();

    layer_gemm<17, 4, 64>(bufA, bufB, Wp0, b0f, lane);   // 68 -> 64
    __syncthreads();
    layer_gemm<16, 4, 64>(bufB, bufA, Wp1, b1f, lane);   // 64 -> 64
    __syncthreads();

    // layer 3 (64 -> 128) fully in registers + max over the 32 samples
    v8f acc[2][8];
    v8f zero = {};
#pragma unroll
    for (int mt = 0; mt < 2; ++mt)
#pragma unroll
        for (int n = 0; n < 8; ++n) acc[mt][n] = zero;

    const int mrow  = lane & 15;
    const int khalf = (lane >> 4) * 2;
    const int ncol  = lane & 15;
    for (int k = 0; k < 16; ++k) {
        const int kc = 4 * k + khalf;
        v2f a0 = *(const v2f*)(bufA + mrow * PITCH + kc);
        v2f a1 = *(const v2f*)(bufA + (mrow + 16) * PITCH + kc);
#pragma unroll
        for (int n = 0; n < 8; ++n) {
            v2f bt = Wp2[(kc >> 1) * 128 + n * 16 + ncol];
            acc[0][n] = wmma4(a0, bt, acc[0][n]);
            acc[1][n] = wmma4(a1, bt, acc[1][n]);
        }
    }
    // max over M (bias shared per-channel & ReLU monotone => pool first, epilogue after)
    float* of = out_feat + (size_t)cg * 128;
#pragma unroll
    for (int n = 0; n < 8; ++n) {
        float r = -3.4e38f;
#pragma unroll
        for (int j = 0; j < 8; ++j) {
            r = fmaxf(r, acc[0][n][j]);
            r = fmaxf(r, acc[1][n][j]);
        }
        r = fmaxf(r, __shfl_xor(r, 16, 32));              // merge lane L <-> L+16
        float v = fmaxf(r + b2f[n * 16 + ncol], 0.0f);
        if (lane < 16) of[n * 16 + ncol] = v;
    }
}

// ---------------------------------------------------------------- host launcher
extern "C" void kernel_launch(void* const* d_in, const int* in_sizes, int n_in,
                              void* d_out, int out_size, void* d_ws, size_t ws_size,
                              hipStream_t stream)
{
    const float* xyz      = (const float*)d_in[0];
    const float* features = (const float*)d_in[1];
    const float* W0 = (const float*)d_in[2];  const float* b0 = (const float*)d_in[3];
    const float* g0 = (const float*)d_in[4];  const float* be0 = (const float*)d_in[5];
    const float* m0 = (const float*)d_in[6];  const float* v0 = (const float*)d_in[7];
    const float* W1 = (const float*)d_in[8];  const float* b1 = (const float*)d_in[9];
    const float* g1 = (const float*)d_in[10]; const float* be1 = (const float*)d_in[11];
    const float* m1 = (const float*)d_in[12]; const float* v1 = (const float*)d_in[13];
    const float* W2 = (const float*)d_in[14]; const float* b2 = (const float*)d_in[15];
    const float* g2 = (const float*)d_in[16]; const float* be2 = (const float*)d_in[17];
    const float* m2 = (const float*)d_in[18]; const float* v2_ = (const float*)d_in[19];

    float* out      = (float*)d_out;
    float* new_xyz  = out;                              // [16,1024,3]
    float* out_feat = out + (size_t)B_ * NPOINT * 3;    // [16,1024,128]

    int*   gidx = (int*)d_ws;                                           // 16*1024*32 ints
    float* wsf  = (float*)((char*)d_ws + (size_t)B_ * NPOINT * NSAMPLE * sizeof(int));
    v2f*   Wp0 = (v2f*)wsf;                  // 34*64 float2
    float* b0f = (float*)(Wp0 + 34 * 64);    // 64
    v2f*   Wp1 = (v2f*)(b0f + 64);           // 32*64 float2
    float* b1f = (float*)(Wp1 + 32 * 64);    // 64
    v2f*   Wp2 = (v2f*)(b1f + 64);           // 32*128 float2
    float* b2f = (float*)(Wp2 + 32 * 128);   // 128

    fuse_weights_kernel<<<1, 256, 0, stream>>>(W0, b0, g0, be0, m0, v0,
                                               W1, b1, g1, be1, m1, v1,
                                               W2, b2, g2, be2, m2, v2_,
                                               Wp0, b0f, Wp1, b1f, Wp2, b2f);
    fps_kernel<<<B_, 256, 0, stream>>>(xyz, new_xyz);
    knn_kernel<<<B_ * 8, KNN_T, 0, stream>>>(xyz, new_xyz, gidx);
    mlp_kernel<<<B_ * NPOINT, 32, 0, stream>>>(xyz, features, gidx, new_xyz, out_feat,
                                               Wp0, b0f, Wp1, b1f, Wp2, b2f);
}